// OptimalTransportLoss_50036368998560
// MI455X (gfx1250) — compile-verified
//
#include <hip/hip_runtime.h>
#include <math.h>

#define NPTS 4096
#define DF   64

typedef float v2f __attribute__((ext_vector_type(2)));
typedef float v8f __attribute__((ext_vector_type(8)));

// ---------------- workspace layout (float offsets) ----------------
static constexpr size_t OFF_X2   = 0;                  // 4096
static constexpr size_t OFF_Y2   = OFF_X2 + NPTS;      // 4096
static constexpr size_t OFF_MIN  = OFF_Y2 + NPTS;      // 64
static constexpr size_t OFF_MAX  = OFF_MIN + DF;       // 64
static constexpr size_t OFF_EPS  = OFF_MAX + DF;       // 73 used
static constexpr size_t OFF_POTA = 8448;               // 4*4096
static constexpr size_t OFF_POTB = OFF_POTA + 4*NPTS;  // 4*4096
static constexpr size_t OFF_FIN  = OFF_POTB + 4*NPTS;  // 4*4096

// sub-buffers within a potential set
static constexpr int P_FBA = 0;
static constexpr int P_GAB = NPTS;
static constexpr int P_FAA = 2*NPTS;
static constexpr int P_GBB = 3*NPTS;

static constexpr float BLUR2   = 0.05f * 0.05f;  // eps_final = blur^p, p=2
static constexpr float DECAY   = 0.9f * 0.9f;    // scaling^p
static constexpr int   NITER   = 72;
static constexpr float INV_LN2 = 1.44269504088896340736f;
static constexpr float LN2     = 0.69314718055994530942f;

struct SmArgs {
  const float* rows;    // points giving matrix rows
  const float* cols;    // points giving matrix cols
  const float* rowsq;   // |row point|^2
  const float* colsq;   // |col point|^2
  const float* g_old;   // potential entering h = log b + g/eps
  const float* pot_old; // potential being damped
  float*       pot_new; // output potential
};
struct SmArgs4 { SmArgs v[4]; };

// ---------------- helper kernels ----------------
__global__ void ot_normsq(const float* __restrict__ x, const float* __restrict__ y,
                          float* __restrict__ x2, float* __restrict__ y2) {
  int i = blockIdx.x * 256 + threadIdx.x;        // 0 .. 8191
  const float* p = (i < NPTS) ? (x + (size_t)i * DF) : (y + (size_t)(i - NPTS) * DF);
  float s = 0.f;
#pragma unroll
  for (int k = 0; k < DF; k += 4) {
    float4 v = *(const float4*)(p + k);
    s += v.x * v.x + v.y * v.y + v.z * v.z + v.w * v.w;
  }
  if (i < NPTS) x2[i] = s; else y2[i - NPTS] = s;
}

__global__ void ot_minmax(const float* __restrict__ x, const float* __restrict__ y,
                          float* __restrict__ mins, float* __restrict__ maxs) {
  __shared__ float smn[128], smx[128];
  int d = blockIdx.x, t = threadIdx.x;
  float mn = __builtin_inff(), mx = -__builtin_inff();
  for (int r = t; r < 2 * NPTS; r += 128) {
    float v = (r < NPTS) ? x[(size_t)r * DF + d] : y[(size_t)(r - NPTS) * DF + d];
    mn = fminf(mn, v); mx = fmaxf(mx, v);
  }
  smn[t] = mn; smx[t] = mx; __syncthreads();
  for (int s = 64; s > 0; s >>= 1) {
    if (t < s) { smn[t] = fminf(smn[t], smn[t + s]); smx[t] = fmaxf(smx[t], smx[t + s]); }
    __syncthreads();
  }
  if (t == 0) { mins[d] = smn[0]; maxs[d] = smx[0]; }
}

__global__ void ot_epslist(const float* __restrict__ mins, const float* __restrict__ maxs,
                           float* __restrict__ epsl) {
  __shared__ float diam2;
  if (threadIdx.x == 0) {
    float s = 0.f;
    for (int d = 0; d < DF; ++d) { float r = maxs[d] - mins[d]; s += r * r; }
    diam2 = s;
  }
  __syncthreads();
  int t = threadIdx.x;
  if (t < NITER) epsl[t] = fmaxf(BLUR2, diam2 * __powf(DECAY, (float)t));
  if (t == NITER) epsl[NITER] = BLUR2;   // final extrapolation eps
}

__global__ void ot_zero(float* __restrict__ p, int n) {
  int i = blockIdx.x * 256 + threadIdx.x;
  if (i < n) p[i] = 0.f;
}

// ---------------- fused WMMA softmin round ----------------
// grid = (NPTS/16, 4); block = 128 (4 waves). blockIdx.y selects softmin variant.
// pot_new[i] = w_old*pot_old[i] + w_new * softmin(eps, C, h),
// h_j = log(1/N) + g_old[j]/eps, C_ij = 0.5(|xi|^2+|yj|^2) - xi.yj via WMMA f32.
// Online logsumexp runs in the log2 domain so v_exp_f32/v_log_f32 are used raw
// (no 1/ln2 pre-multiplies in the hot loop); log2(1/4096) = -12 exactly.
__global__ __launch_bounds__(128) void ot_softmin_round(
    SmArgs4 A4, const float* __restrict__ eps_list, int eps_idx,
    float w_old, float w_new)
{
  const SmArgs a = A4.v[blockIdx.y];
  const int lane = threadIdx.x & 31;
  const int wave = threadIdx.x >> 5;
  const int lh   = lane & 15;     // M index for A, N index for B/C (ISA layout)
  const int kh   = lane >> 4;     // K-half selector for A/B fragments
  const int i0   = blockIdx.x * 16;

  const float eps      = eps_list[eps_idx];
  const float inv_eps  = 1.0f / eps;
  const float inv_eps2 = inv_eps * INV_LN2;   // 1/(eps*ln2)
  const float logb2    = -12.0f;              // log2(1/4096)

  // A fragments: 16x4 f32 per k-step, 16 steps cover K=64; pinned in registers.
  // Layout (ISA 7.12.2): lane lh holds row M=lh; VGPR v holds K = k0 + 2*kh + v.
  v2f afrag[16];
  const float* arow = a.rows + (size_t)(i0 + lh) * DF + 2 * kh;
#pragma unroll
  for (int ks = 0; ks < 16; ++ks) afrag[ks] = *(const v2f*)(arow + ks * 4);

  // online log2-sum-exp2 state per C-register (row m = r + 8*half)
  float mrun[8], srun[8];
#pragma unroll
  for (int r = 0; r < 8; ++r) { mrun[r] = -__builtin_inff(); srun[r] = 0.0f; }

  const int jbeg = wave * (NPTS / 4);
  const int jend = jbeg + (NPTS / 4);
  for (int j0 = jbeg; j0 < jend; j0 += 16) {
    // speculative prefetch of the next column tile (dropped if OOB)
    __builtin_prefetch(a.cols + (size_t)(j0 + 16 + lh) * DF, 0, 3);

    // per-column constant (log2 domain): (h_j - 0.5|yj|^2/eps)/ln2
    const float gv  = a.g_old[j0 + lh];
    const float q   = a.colsq[j0 + lh];
    const float hy2 = logb2 + (gv - 0.5f * q) * inv_eps2;

    // preload ALL B fragments first -> loads issue as one clause, then the
    // 16 V_WMMA_F32_16X16X4_F32 run back-to-back behind a single wait
    const float* brow = a.cols + (size_t)(j0 + lh) * DF + 2 * kh;
    v2f bfrag[16];
#pragma unroll
    for (int ks = 0; ks < 16; ++ks) bfrag[ks] = *(const v2f*)(brow + ks * 4);

    v8f c = {0.f, 0.f, 0.f, 0.f, 0.f, 0.f, 0.f, 0.f};
#pragma unroll
    for (int ks = 0; ks < 16; ++ks) {
      c = __builtin_amdgcn_wmma_f32_16x16x4_f32(
            /*neg_a=*/false, afrag[ks], /*neg_b=*/false, bfrag[ks],
            /*c_mod=*/(short)0, c, /*reuse_a=*/false, /*reuse_b=*/false);
    }

    // s2_ij = (h_j - C_ij/eps)/ln2  (row-constant 0.5|xi|^2/eps hoisted out)
#pragma unroll
    for (int r = 0; r < 8; ++r) {
      float s  = fmaf(c[r], inv_eps2, hy2);
      float nm = fmaxf(mrun[r], s);
      srun[r]  = srun[r] * __builtin_amdgcn_exp2f(mrun[r] - nm)
               + __builtin_amdgcn_exp2f(s - nm);
      mrun[r]  = nm;
    }
  }

  // combine the 16 lane-partials of each row within each 16-lane half
  __shared__ float shm[4][16];
  __shared__ float shs[4][16];
#pragma unroll
  for (int r = 0; r < 8; ++r) {
    float mv = mrun[r], sv = srun[r];
#pragma unroll
    for (int off = 1; off < 16; off <<= 1) {
      float mo = __shfl_xor(mv, off, 32);
      float so = __shfl_xor(sv, off, 32);
      float nm = fmaxf(mv, mo);
      sv = sv * __builtin_amdgcn_exp2f(mv - nm) + so * __builtin_amdgcn_exp2f(mo - nm);
      mv = nm;
    }
    if (lh == r) { shm[wave][8 * kh + r] = mv; shs[wave][8 * kh + r] = sv; }
  }
  __syncthreads();

  // merge 4 wave partials per row, finalize, damp, store
  if (threadIdx.x < 16) {
    int rl = threadIdx.x;
    float mv = shm[0][rl], sv = shs[0][rl];
#pragma unroll
    for (int w = 1; w < 4; ++w) {
      float mo = shm[w][rl], so = shs[w][rl];
      float nm = fmaxf(mv, mo);
      sv = sv * __builtin_amdgcn_exp2f(mv - nm) + so * __builtin_amdgcn_exp2f(mo - nm);
      mv = nm;
    }
    float lse = LN2 * (mv + __builtin_amdgcn_logf(sv));  // back to natural log
    int row   = i0 + rl;
    float val = 0.5f * a.rowsq[row] - eps * lse;         // softmin value
    a.pot_new[row] = w_old * a.pot_old[row] + w_new * val;
  }
}

// ---------------- deterministic final reduction ----------------
__global__ void ot_reduce(const float* __restrict__ fin, float* __restrict__ out) {
  __shared__ float sd[256];
  float acc = 0.f;
  for (int i = threadIdx.x; i < NPTS; i += 256)
    acc += (fin[P_FBA + i] - fin[P_FAA + i]) + (fin[P_GAB + i] - fin[P_GBB + i]);
  sd[threadIdx.x] = acc;
  __syncthreads();
  for (int s = 128; s > 0; s >>= 1) {
    if (threadIdx.x < s) sd[threadIdx.x] += sd[threadIdx.x + s];
    __syncthreads();
  }
  if (threadIdx.x == 0) out[0] = sd[0] * (1.0f / NPTS);
}

// ---------------- host side ----------------
static SmArgs4 make_args(const float* x, const float* y,
                         const float* x2, const float* y2,
                         float* oldb, float* newb) {
  SmArgs4 a;
  // ft : rows=x cols=y, h from g_ab, damps f_ba
  a.v[0] = { x, y, x2, y2, oldb + P_GAB, oldb + P_FBA, newb + P_FBA };
  // gt : rows=y cols=x, h from f_ba, damps g_ab
  a.v[1] = { y, x, y2, x2, oldb + P_FBA, oldb + P_GAB, newb + P_GAB };
  // faa: rows=x cols=x, h from f_aa, damps f_aa
  a.v[2] = { x, x, x2, x2, oldb + P_FAA, oldb + P_FAA, newb + P_FAA };
  // gbb: rows=y cols=y, h from g_bb, damps g_bb
  a.v[3] = { y, y, y2, y2, oldb + P_GBB, oldb + P_GBB, newb + P_GBB };
  return a;
}

extern "C" void kernel_launch(void* const* d_in, const int* in_sizes, int n_in,
                              void* d_out, int out_size, void* d_ws, size_t ws_size,
                              hipStream_t stream) {
  const float* x = (const float*)d_in[0];
  const float* y = (const float*)d_in[1];
  float* ws   = (float*)d_ws;
  float* x2   = ws + OFF_X2;
  float* y2   = ws + OFF_Y2;
  float* mins = ws + OFF_MIN;
  float* maxs = ws + OFF_MAX;
  float* epsl = ws + OFF_EPS;
  float* potA = ws + OFF_POTA;
  float* potB = ws + OFF_POTB;
  float* fin  = ws + OFF_FIN;

  ot_normsq<<<dim3((2 * NPTS) / 256), 256, 0, stream>>>(x, y, x2, y2);
  ot_minmax<<<dim3(DF), 128, 0, stream>>>(x, y, mins, maxs);
  ot_epslist<<<1, 128, 0, stream>>>(mins, maxs, epsl);
  ot_zero<<<dim3((8 * NPTS) / 256), 256, 0, stream>>>(potA, 8 * NPTS); // zero potA+potB

  dim3 grid(NPTS / 16, 4), blk(128);

  // init: potentials from softmin(eps0, C, log b) — g_old buffers are zero
  ot_softmin_round<<<grid, blk, 0, stream>>>(make_args(x, y, x2, y2, potA, potB),
                                             epsl, 0, 0.0f, 1.0f);
  float* cur = potB; float* oth = potA;
  for (int t = 0; t < NITER; ++t) {   // 72 damped symmetric updates, ping-pong
    ot_softmin_round<<<grid, blk, 0, stream>>>(make_args(x, y, x2, y2, cur, oth),
                                               epsl, t, 0.5f, 0.5f);
    float* tmp = cur; cur = oth; oth = tmp;
  }
  // final extrapolation at eps = blur^2 (eps_list[72]), no damping
  ot_softmin_round<<<grid, blk, 0, stream>>>(make_args(x, y, x2, y2, cur, fin),
                                             epsl, NITER, 0.0f, 1.0f);

  ot_reduce<<<1, 256, 0, stream>>>(fin, (float*)d_out);
}